// GATEncoder_81604378624632
// MI455X (gfx1250) — compile-verified
//
#include <hip/hip_runtime.h>
#include <math.h>

// ---------------------------------------------------------------------------
// Types for CDNA5 WMMA (fp32 16x16x4): A/B = 2 VGPRs/lane, C/D = 8 VGPRs/lane
// ---------------------------------------------------------------------------
typedef __attribute__((ext_vector_type(2))) float v2f;
typedef __attribute__((ext_vector_type(8))) float v8f;

#define TS   32   // output tile (M and N) per 128-thread block (2x2 waves)
#define KC   32   // K chunk staged in LDS
#define ASTR 36   // padded LDS row stride for A tile
#define BSTR 33   // padded LDS row stride for B tile

static __device__ __forceinline__ float leaky02(float x) {
    return x > 0.f ? x : 0.2f * x;
}

// float atomic max via int/uint bit tricks (correct for mixed signs, -inf init)
static __device__ __forceinline__ void atomicMaxF(float* addr, float v) {
    if (v >= 0.f)
        atomicMax((int*)addr, __float_as_int(v));
    else
        atomicMin((unsigned int*)addr, __float_as_uint(v));
}

// ---------------------------------------------------------------------------
// fp32 GEMM via V_WMMA_F32_16X16X4_F32.  D[M,N] = A[M,K] * B[K,N] (row major)
// grid = (N/32, M/32), block = 128 threads (4 waves, 2x2 wave tile grid).
// Requires M,N,K multiples of 32 (true here: 16384 / 512 / 256).
// ---------------------------------------------------------------------------
__global__ __launch_bounds__(128)
void gemm_wmma_f32(const float* __restrict__ A, const float* __restrict__ B,
                   float* __restrict__ D, int M, int K, int N) {
    __shared__ float As[TS * ASTR];
    __shared__ float Bs[KC * BSTR];

    const int tileN = blockIdx.x * TS;
    const int tileM = blockIdx.y * TS;
    const int t    = threadIdx.x;
    const int wv   = t >> 5;          // wave id 0..3
    const int lane = t & 31;
    const int half = lane >> 4;       // 0: lanes 0-15, 1: lanes 16-31
    const int lm   = lane & 15;
    const int wm   = wv >> 1;         // wave row 0..1
    const int wn   = wv & 1;          // wave col 0..1

    // LDS staging: each of 128 threads copies 8 floats of A and 8 of B
    const int crow = t >> 2;          // 0..31
    const int ccol = (t & 3) * 8;     // 0,8,16,24

    v8f acc = {};

    for (int kt = 0; kt < K; kt += KC) {
        const float* ga = A + (size_t)(tileM + crow) * K + kt + ccol;
        const float* gb = B + (size_t)(kt + crow) * N + tileN + ccol;
        if (kt + KC < K) {
            __builtin_prefetch(ga + KC, 0, 0);               // global_prefetch_b8
            __builtin_prefetch(gb + (size_t)KC * N, 0, 0);
        }
        #pragma unroll
        for (int j = 0; j < 8; ++j) As[crow * ASTR + ccol + j] = ga[j];
        #pragma unroll
        for (int j = 0; j < 8; ++j) Bs[crow * BSTR + ccol + j] = gb[j];
        __syncthreads();

        #pragma unroll
        for (int kk = 0; kk < KC; kk += 4) {
            // ISA layout: A lane holds M=lane%16, K = {2*(lane/16), +1}
            //             B lane holds N=lane%16, K = {2*(lane/16), +1}
            const int k0 = kk + 2 * half;
            v2f a, b;
            a.x = As[(wm * 16 + lm) * ASTR + k0];
            a.y = As[(wm * 16 + lm) * ASTR + k0 + 1];
            b.x = Bs[(k0)     * BSTR + wn * 16 + lm];
            b.y = Bs[(k0 + 1) * BSTR + wn * 16 + lm];
            acc = __builtin_amdgcn_wmma_f32_16x16x4_f32(
                false, a, false, b, (short)0, acc, false, false);
        }
        __syncthreads();
    }

    // D layout: VGPR r -> row r (lanes 0-15) / r+8 (lanes 16-31), col = lane%16
    const int col   = tileN + wn * 16 + lm;
    const int rbase = tileM + wm * 16 + half * 8;
    #pragma unroll
    for (int r = 0; r < 8; ++r)
        D[(size_t)(rbase + r) * N + col] = acc[r];
}

// ---------------------------------------------------------------------------
// Per-node attention logits: al_s[n,h] = sum_c h[n,h,c]*a_s[h,c]  (same al_d)
// ---------------------------------------------------------------------------
__global__ void att_logits(const float* __restrict__ h,
                           const float* __restrict__ a_s,
                           const float* __restrict__ a_d,
                           float* __restrict__ al_s, float* __restrict__ al_d,
                           int N, int H, int C) {
    int t = blockIdx.x * blockDim.x + threadIdx.x;
    if (t >= N * H) return;
    int n = t / H, hh = t - n * H;
    const float* row = h + (size_t)n * H * C + (size_t)hh * C;
    const float* vs  = a_s + (size_t)hh * C;
    const float* vd  = a_d + (size_t)hh * C;
    float ss = 0.f, sd = 0.f;
    for (int c = 0; c < C; ++c) {
        float v = row[c];
        ss = fmaf(v, vs[c], ss);
        sd = fmaf(v, vd[c], sd);
    }
    al_s[t] = ss;
    al_d[t] = sd;
}

__global__ void fill_f32(float* __restrict__ p, float v, size_t n) {
    size_t i = blockIdx.x * (size_t)blockDim.x + threadIdx.x;
    if (i < n) p[i] = v;
}

// ---------------------------------------------------------------------------
// Edge passes (with implicit self loops: e in [E, E+N) -> src=dst=e-E)
// ---------------------------------------------------------------------------
__global__ void edge_max(const int* __restrict__ ei, int E, int N, int H,
                         const float* __restrict__ al_s,
                         const float* __restrict__ al_d,
                         float* __restrict__ mbuf) {
    int t = blockIdx.x * blockDim.x + threadIdx.x;
    int tot = (E + N) * H;
    if (t >= tot) return;
    int e = t / H, hh = t - e * H;
    int s, d;
    if (e < E) { s = ei[e]; d = ei[E + e]; } else { s = d = e - E; }
    float v = leaky02(al_s[s * H + hh] + al_d[d * H + hh]);
    atomicMaxF(&mbuf[d * H + hh], v);
}

__global__ void edge_exp(const int* __restrict__ ei, int E, int N, int H,
                         const float* __restrict__ al_s,
                         const float* __restrict__ al_d,
                         const float* __restrict__ mbuf,
                         float* __restrict__ exbuf,
                         float* __restrict__ denom) {
    int t = blockIdx.x * blockDim.x + threadIdx.x;
    int tot = (E + N) * H;
    if (t >= tot) return;
    int e = t / H, hh = t - e * H;
    int s, d;
    if (e < E) { s = ei[e]; d = ei[E + e]; } else { s = d = e - E; }
    float v  = leaky02(al_s[s * H + hh] + al_d[d * H + hh]);
    float ex = __expf(v - mbuf[d * H + hh]);
    exbuf[t] = ex;
    atomicAdd(&denom[d * H + hh], ex);
}

// agg[dst, f] += h[src, f] * alpha(e, f/C); one thread per (edge, feature)
__global__ void edge_aggregate(const int* __restrict__ ei, int E, int N,
                               int H, int C,
                               const float* __restrict__ hfeat,
                               const float* __restrict__ exbuf,
                               const float* __restrict__ denom,
                               float* __restrict__ agg) {
    const int F = H * C;
    long long t = blockIdx.x * (long long)blockDim.x + threadIdx.x;
    long long tot = (long long)(E + N) * F;
    if (t >= tot) return;
    int e = (int)(t / F);
    int f = (int)(t - (long long)e * F);
    int hh = f / C;
    int s, d;
    if (e < E) { s = ei[e]; d = ei[E + e]; } else { s = d = e - E; }
    float alpha = exbuf[(size_t)e * H + hh] / denom[(size_t)d * H + hh];
    atomicAdd(&agg[(size_t)d * F + f], hfeat[(size_t)s * F + f] * alpha);
}

__global__ void bias_relu_inplace(float* __restrict__ p,
                                  const float* __restrict__ b,
                                  size_t total, int F) {
    size_t t = blockIdx.x * (size_t)blockDim.x + threadIdx.x;
    if (t >= total) return;
    int f = (int)(t % F);
    float v = p[t] + b[f];
    p[t] = v > 0.f ? v : 0.f;
}

__global__ void mean_heads_bias(const float* __restrict__ agg,
                                const float* __restrict__ b,
                                float* __restrict__ out, int N, int H, int C) {
    int t = blockIdx.x * blockDim.x + threadIdx.x;
    if (t >= N * C) return;
    int n = t / C, c = t - n * C;
    float s = 0.f;
    for (int hh = 0; hh < H; ++hh)
        s += agg[(size_t)n * H * C + (size_t)hh * C + c];
    out[t] = s / (float)H + b[c];
}

// ---------------------------------------------------------------------------
static inline unsigned cdiv_u(size_t a, unsigned b) {
    return (unsigned)((a + b - 1) / b);
}

extern "C" void kernel_launch(void* const* d_in, const int* in_sizes, int n_in,
                              void* d_out, int out_size, void* d_ws, size_t ws_size,
                              hipStream_t stream) {
    const float* x     = (const float*)d_in[0];
    const int*   ei    = (const int*)d_in[1];
    const float* W1    = (const float*)d_in[2];
    const float* a_s1  = (const float*)d_in[3];
    const float* a_d1  = (const float*)d_in[4];
    const float* b1    = (const float*)d_in[5];
    const float* W2    = (const float*)d_in[6];
    const float* a_s2  = (const float*)d_in[7];
    const float* a_d2  = (const float*)d_in[8];
    const float* b2    = (const float*)d_in[9];
    float*       out   = (float*)d_out;

    // Derive dimensions from input sizes
    const int F1   = in_sizes[3];              // H*C1 = 512
    const int F2   = in_sizes[7];              // H*LAT = 256
    const int LAT  = in_sizes[9];              // 64
    const int H    = F2 / LAT;                 // 4
    const int C1   = F1 / H;                   // 128
    const int Fin  = in_sizes[2] / F1;         // 512
    const int N    = in_sizes[0] / Fin;        // 16384
    const int E    = in_sizes[1] / 2;          // 262144
    const int Etot = E + N;

    // Workspace layout (floats)
    float* ws = (float*)d_ws;
    size_t o = 0;
    float* h1   = ws + o; o += (size_t)N * F1;       // 32 MB; reused in layer 2
    float* agg1 = ws + o; o += (size_t)N * F1;       // 32 MB
    float* ex   = ws + o; o += (size_t)Etot * H;
    float* als  = ws + o; o += (size_t)N * H;
    float* ald  = ws + o; o += (size_t)N * H;
    float* mbuf = ws + o; o += (size_t)N * H;
    float* den  = ws + o; o += (size_t)N * H;
    // Layer 2 reuses the h1 slab: h2 (N*F2) + agg2 (N*F2) == N*F1 floats
    float* h2   = h1;
    float* agg2 = h1 + (size_t)N * F2;
    (void)ws_size; (void)n_in; (void)out_size;

    const unsigned TB = 256;
    const float NEG_INF = -__builtin_huge_valf();

    // ======================= Layer 1 =======================
    fill_f32<<<cdiv_u((size_t)N * F1, TB), TB, 0, stream>>>(agg1, 0.f, (size_t)N * F1);
    fill_f32<<<cdiv_u((size_t)N * H, TB), TB, 0, stream>>>(mbuf, NEG_INF, (size_t)N * H);
    fill_f32<<<cdiv_u((size_t)N * H, TB), TB, 0, stream>>>(den, 0.f, (size_t)N * H);

    gemm_wmma_f32<<<dim3(F1 / TS, N / TS), 128, 0, stream>>>(x, W1, h1, N, Fin, F1);

    att_logits<<<cdiv_u((size_t)N * H, TB), TB, 0, stream>>>(h1, a_s1, a_d1, als, ald, N, H, C1);

    edge_max<<<cdiv_u((size_t)Etot * H, TB), TB, 0, stream>>>(ei, E, N, H, als, ald, mbuf);
    edge_exp<<<cdiv_u((size_t)Etot * H, TB), TB, 0, stream>>>(ei, E, N, H, als, ald, mbuf, ex, den);
    edge_aggregate<<<cdiv_u((size_t)Etot * F1, TB), TB, 0, stream>>>(
        ei, E, N, H, C1, h1, ex, den, agg1);

    bias_relu_inplace<<<cdiv_u((size_t)N * F1, TB), TB, 0, stream>>>(agg1, b1, (size_t)N * F1, F1);

    // ======================= Layer 2 =======================
    gemm_wmma_f32<<<dim3(F2 / TS, N / TS), 128, 0, stream>>>(agg1, W2, h2, N, F1, F2);

    fill_f32<<<cdiv_u((size_t)N * F2, TB), TB, 0, stream>>>(agg2, 0.f, (size_t)N * F2);
    fill_f32<<<cdiv_u((size_t)N * H, TB), TB, 0, stream>>>(mbuf, NEG_INF, (size_t)N * H);
    fill_f32<<<cdiv_u((size_t)N * H, TB), TB, 0, stream>>>(den, 0.f, (size_t)N * H);

    att_logits<<<cdiv_u((size_t)N * H, TB), TB, 0, stream>>>(h2, a_s2, a_d2, als, ald, N, H, LAT);

    edge_max<<<cdiv_u((size_t)Etot * H, TB), TB, 0, stream>>>(ei, E, N, H, als, ald, mbuf);
    edge_exp<<<cdiv_u((size_t)Etot * H, TB), TB, 0, stream>>>(ei, E, N, H, als, ald, mbuf, ex, den);
    edge_aggregate<<<cdiv_u((size_t)Etot * F2, TB), TB, 0, stream>>>(
        ei, E, N, H, LAT, h2, ex, den, agg2);

    mean_heads_bias<<<cdiv_u((size_t)N * LAT, TB), TB, 0, stream>>>(agg2, b2, out, N, H, LAT);
}